// CustomFullyConnectedLayer_14345190768667
// MI455X (gfx1250) — compile-verified
//
#include <hip/hip_runtime.h>
#include <cstdint>

// Problem constants (from reference)
#define NDIM   3072
#define BATCHN 8192
#define KTOP   16.0f
#define INV_L  100.0f      // 1 / ALPHA_LR
#define NITER  50
#define MAXD   512         // cap on retained diagonals (actual ~32)

#define PREP_T 256
#define EPT    12          // NDIM / PREP_T

#define MT 256             // main kernel block size
#define BT 8               // batch rows per block

// ---------------------------------------------------------------------------
// Main kernel: banded (pseudo-diagonal) matmul.
//   out[b, r] = sum_j val[j] * V[idx[j], (r - idx[j]) mod n] * x[b, (r - idx[j]) mod n]
// Block = 8 batch rows x all 3072 outputs. The 96 KB x tile is staged to LDS
// with CDNA5 async global->LDS b128 copies (ASYNCcnt-tracked); each scaled V
// element is fetched once per (r, diag) from L2-resident rows and reused
// across the 8 batch rows from LDS.
// ---------------------------------------------------------------------------
__global__ __launch_bounds__(MT)
void diag_mm_kernel(const float* __restrict__ xin,
                    const float* __restrict__ V,
                    const int*   __restrict__ nnz_p,
                    const int*   __restrict__ idx_p,
                    const float* __restrict__ val_p,
                    float* __restrict__ out)
{
    __shared__ float xs[BT * NDIM];     // 96 KB x tile (bb-major, rows contiguous)
    __shared__ int   sidx[MAXD];
    __shared__ float sval[MAXD];

    const int tid = threadIdx.x;
    const int b0  = blockIdx.x * BT;
    const int nnz = *nnz_p;             // uniform scalar

    // stage diagonal table to LDS
    for (int j = tid; j < nnz; j += MT) { sidx[j] = idx_p[j]; sval[j] = val_p[j]; }

    // async-copy 8 contiguous x rows (96 KB) into LDS: 24 x b128 per thread
    const float* xg = xin + (size_t)b0 * NDIM;          // uniform base -> SGPR pair
    const uint32_t lds_base = (uint32_t)(uintptr_t)(&xs[0]);
#pragma unroll
    for (int k2 = 0; k2 < (BT * NDIM * 4) / (16 * MT); ++k2) {
        const uint32_t boff = (uint32_t)((k2 * MT + tid) * 16);
        asm volatile("global_load_async_to_lds_b128 %0, %1, %2"
                     :
                     : "v"(lds_base + boff), "v"(boff), "s"(xg)
                     : "memory");
    }
    asm volatile("s_wait_asynccnt 0" ::: "memory");
    __syncthreads();

#pragma unroll 1
    for (int rc = 0; rc < NDIM / MT; ++rc) {
        const int r = rc * MT + tid;
        float acc[BT];
#pragma unroll
        for (int bb = 0; bb < BT; ++bb) acc[bb] = 0.0f;

#pragma unroll 2
        for (int j = 0; j < nnz; ++j) {
            const int i = sidx[j];
            int c = r - i; if (c < 0) c += NDIM;           // c = (r - i) mod n
            const float w = sval[j] * V[i * NDIM + c];     // L2-resident rows
#pragma unroll
            for (int bb = 0; bb < BT; ++bb)
                acc[bb] = fmaf(xs[bb * NDIM + c], w, acc[bb]);
        }
#pragma unroll
        for (int bb = 0; bb < BT; ++bb)
            out[(size_t)(b0 + bb) * NDIM + r] = acc[bb];   // coalesced per bb
    }
}

// ---------------------------------------------------------------------------
// Prep kernel: Dykstra soft top-k projection + deterministic compaction.
// One workgroup. Per-element state (x, q) lives in registers; the hyperplane
// residual p is provably a uniform scalar, so each iteration needs exactly one
// block-wide sum (wave32 shfl reduce + tiny LDS combine).
// ---------------------------------------------------------------------------
__global__ __launch_bounds__(PREP_T)
void prep_kernel(const float* __restrict__ alpha,
                 int* __restrict__ nnz_out,
                 int* __restrict__ idx_out,
                 float* __restrict__ val_out)
{
    __shared__ float red[PREP_T / 32];
    __shared__ int   cnt[PREP_T];
    __shared__ int   offs[PREP_T];

    const int tid  = threadIdx.x;
    const int base = tid * EPT;            // contiguous ownership -> sorted output

    float x[EPT], q[EPT];
#pragma unroll
    for (int j = 0; j < EPT; ++j) {
        x[j] = alpha[base + j] * INV_L;    // z = alpha / l
        q[j] = 0.0f;
    }
    float p = 0.0f;
    const float invn = 1.0f / (float)NDIM;

    for (int it = 0; it < NITER; ++it) {
        // block-wide sum of x
        float s = 0.0f;
#pragma unroll
        for (int j = 0; j < EPT; ++j) s += x[j];
        for (int m = 16; m >= 1; m >>= 1) s += __shfl_xor(s, m, 32);
        const int wave = tid >> 5, lane = tid & 31;
        __syncthreads();                    // protect red[] WAR across iterations
        if (lane == 0) red[wave] = s;
        __syncthreads();
        float S = 0.0f;
#pragma unroll
        for (int w = 0; w < PREP_T / 32; ++w) S += red[w];

        // hyperplane projection of (x + p): uniform shift t
        const float t  = (KTOP - (S + (float)NDIM * p)) * invn;
        const float sh = p + t;
        // box projection of (y' + q); clip produces exact zeros
#pragma unroll
        for (int j = 0; j < EPT; ++j) {
            const float e  = x[j] + sh + q[j];
            const float xc = fminf(fmaxf(e, 0.0f), 1.0f);
            q[j] = e - xc;
            x[j] = xc;
        }
        p = -t;
    }

    // deterministic compaction of nonzeros (serial scan -> stable order)
    int c = 0;
#pragma unroll
    for (int j = 0; j < EPT; ++j) c += (x[j] > 0.0f) ? 1 : 0;
    cnt[tid] = c;
    __syncthreads();
    if (tid == 0) {
        int run = 0;
        for (int t2 = 0; t2 < PREP_T; ++t2) { offs[t2] = run; run += cnt[t2]; }
        *nnz_out = (run > MAXD) ? MAXD : run;
    }
    __syncthreads();
    int o = offs[tid];
#pragma unroll
    for (int j = 0; j < EPT; ++j) {
        const float v = x[j];
        if (v > 0.0f) {
            if (o < MAXD) { idx_out[o] = base + j; val_out[o] = v; }
            ++o;
        }
    }
}

// ---------------------------------------------------------------------------
extern "C" void kernel_launch(void* const* d_in, const int* in_sizes, int n_in,
                              void* d_out, int out_size, void* d_ws, size_t ws_size,
                              hipStream_t stream)
{
    const float* x     = (const float*)d_in[0];   // (8192, 3072)
    const float* V     = (const float*)d_in[1];   // (3072, 3072)
    const float* alpha = (const float*)d_in[2];   // (3072,)
    float* out = (float*)d_out;                   // (8192, 3072) f32

    // workspace layout: [nnz:int][pad][idx:int*MAXD][val:float*MAXD]
    int*   nnz = (int*)d_ws;
    int*   idx = (int*)((char*)d_ws + 16);
    float* val = (float*)((char*)d_ws + 16 + MAXD * sizeof(int));

    prep_kernel<<<1, PREP_T, 0, stream>>>(alpha, nnz, idx, val);
    diag_mm_kernel<<<BATCHN / BT, MT, 0, stream>>>(x, V, nnz, idx, val, out);
}